// OriginCaps_24180665876666
// MI455X (gfx1250) — compile-verified
//
#include <hip/hip_runtime.h>
#include <math.h>

typedef __attribute__((ext_vector_type(2))) float v2f;
typedef __attribute__((ext_vector_type(4))) float v4f;
typedef __attribute__((ext_vector_type(8))) float v8f;

#define B_SZ  32
#define NCAPS 32
#define ICAPS 2048
#define DCAPS 16
#define IDIM  128

// ---------------------------------------------------------------------------
// Phase 1: u_hat[b,n,c,d] = sum_i W[n,c,d,i] * x[b,c,i]
// Per (n,c): D[16(d) x 32(b)] = A(W tile 16x128) * B(x^T 128x32)
// via V_WMMA_F32_16X16X4_F32, two N-tiles sharing the A fragment.
// Grid: (ICAPS/64, NCAPS); block = 256 threads = 8 waves; wave handles 8 c's.
// ---------------------------------------------------------------------------
__global__ __launch_bounds__(256) void uhat_wmma_kernel(
    const float* __restrict__ x,      // [32, 2048, 128]
    const float* __restrict__ W,      // [32, 2048, 16, 128]
    float* __restrict__ uhat)         // [32, 32, 2048, 16]
{
  const int n      = blockIdx.y;
  const int wave   = threadIdx.x >> 5;
  const int lane   = threadIdx.x & 31;
  const int lane16 = lane & 15;
  const int hi     = lane >> 4;            // 0: lanes 0-15 (K0/K1), 1: lanes 16-31 (K2/K3)
  const int c0     = blockIdx.x * 64 + wave * 8;

  for (int j = 0; j < 8; ++j) {
    const int c = c0 + j;
    v8f acc0 = {0.f,0.f,0.f,0.f,0.f,0.f,0.f,0.f};
    v8f acc1 = {0.f,0.f,0.f,0.f,0.f,0.f,0.f,0.f};

    // A fragment source: W[n, c, m=lane16, k + 2*hi + {0,1}]
    const float* wp  = W + (((size_t)n * ICAPS + c) * DCAPS + lane16) * IDIM + (hi ? 2 : 0);
    // B fragment source: x[b=lane16(+16), c, k + 2*hi + {0,1}]
    const float* xp0 = x + ((size_t)lane16 * ICAPS + c) * IDIM + (hi ? 2 : 0);
    const float* xp1 = xp0 + (size_t)16 * ICAPS * IDIM;

#pragma unroll 8
    for (int k = 0; k < IDIM; k += 4) {
      v2f a  = { wp[k],  wp[k + 1]  };
      v2f b0 = { xp0[k], xp0[k + 1] };
      v2f b1 = { xp1[k], xp1[k + 1] };
      acc0 = __builtin_amdgcn_wmma_f32_16x16x4_f32(false, a, false, b0,
                                                   (short)0, acc0, false, false);
      acc1 = __builtin_amdgcn_wmma_f32_16x16x4_f32(false, a, false, b1,
                                                   (short)0, acc1, false, false);
    }

    // D layout: VGPR j -> d = j (+8 for hi lanes), N = lane16 -> batch.
    const int dbase = hi ? 8 : 0;
    float* up0 = uhat + (((size_t)lane16        * NCAPS + n) * ICAPS + c) * DCAPS + dbase;
    float* up1 = uhat + (((size_t)(lane16 + 16) * NCAPS + n) * ICAPS + c) * DCAPS + dbase;
    v4f l0 = { acc0[0], acc0[1], acc0[2], acc0[3] };
    v4f h0 = { acc0[4], acc0[5], acc0[6], acc0[7] };
    v4f l1 = { acc1[0], acc1[1], acc1[2], acc1[3] };
    v4f h1 = { acc1[4], acc1[5], acc1[6], acc1[7] };
    *(v4f*)(up0)     = l0;
    *(v4f*)(up0 + 4) = h0;
    *(v4f*)(up1)     = l1;
    *(v4f*)(up1 + 4) = h1;
  }
}

// ---------------------------------------------------------------------------
// Phase 2: dynamic routing. One block per batch element b.
// 1024 threads = 32 waves; wave w owns output capsule n = w (wave32-native).
// Routing logits [32][2048] live entirely in LDS (256 KB of the 320 KB WGP LDS).
// ---------------------------------------------------------------------------
__global__ __launch_bounds__(1024) void routing_kernel(
    const float* __restrict__ uhat,   // [32, 32, 2048, 16]
    float* __restrict__ out_v,        // [32, 32, 16]
    float* __restrict__ out_p)        // [32, 32]
{
  extern __shared__ float smem[];
  float* Lg     = smem;                    // [32][2048] routing logits
  float* mx     = Lg + NCAPS * ICAPS;      // [2048] per-c column max
  float* invden = mx + ICAPS;              // [2048] per-c 1/sum(exp)
  float* sS     = invden + ICAPS;          // [32][16] s
  float* sV     = sS + NCAPS * DCAPS;      // [32][16] v
  float* sP     = sV + NCAPS * DCAPS;      // [32]     scale (p)

  const int b    = blockIdx.x;
  const int tid  = threadIdx.x;
  const int n    = tid >> 5;               // wave id == capsule id
  const int lane = tid & 31;
  const float* ub = uhat + ((size_t)b * NCAPS + n) * ICAPS * DCAPS;

  for (int i = tid; i < NCAPS * ICAPS; i += 1024) Lg[i] = 0.0f;
  __syncthreads();

  for (int iter = 0; iter < 3; ++iter) {
    // ---- softmax stats over n for each in_cap c ----
    for (int c = tid; c < ICAPS; c += 1024) {
      float m = -3.4e38f;
      for (int k = 0; k < NCAPS; ++k) m = fmaxf(m, Lg[k * ICAPS + c]);
      float s = 0.0f;
      for (int k = 0; k < NCAPS; ++k) s += __expf(Lg[k * ICAPS + c] - m);
      mx[c] = m;
      invden[c] = 1.0f / s;
    }
    __syncthreads();

    // ---- s[n,:] = sum_c softmax(b)[n,c] * u_hat[b,n,c,:] ----
    float acc[DCAPS];
#pragma unroll
    for (int d = 0; d < DCAPS; ++d) acc[d] = 0.0f;

    for (int c = lane; c < ICAPS; c += 32) {
      float cc = __expf(Lg[n * ICAPS + c] - mx[c]) * invden[c];
      const v4f* up = (const v4f*)(ub + (size_t)c * DCAPS);
      v4f q0 = up[0], q1 = up[1], q2 = up[2], q3 = up[3];
#pragma unroll
      for (int d = 0; d < 4; ++d) {
        acc[d]      += cc * q0[d];
        acc[d + 4]  += cc * q1[d];
        acc[d + 8]  += cc * q2[d];
        acc[d + 12] += cc * q3[d];
      }
    }
    // wave32 butterfly reduction across the 32 lanes of capsule n
#pragma unroll
    for (int off = 16; off > 0; off >>= 1) {
#pragma unroll
      for (int d = 0; d < DCAPS; ++d)
        acc[d] += __shfl_xor(acc[d], off, 32);
    }
    if (lane == 0) {
#pragma unroll
      for (int d = 0; d < DCAPS; ++d) sS[n * DCAPS + d] = acc[d];
    }
    __syncthreads();

    // ---- squash: one thread per capsule ----
    if (tid < NCAPS) {
      float sq = 0.0f;
#pragma unroll
      for (int d = 0; d < DCAPS; ++d) {
        float t = sS[tid * DCAPS + d];
        sq += t * t;
      }
      float scale = sq / (1.0f + sq);
      float inv   = scale * rsqrtf(sq + 1e-11f);
#pragma unroll
      for (int d = 0; d < DCAPS; ++d) sV[tid * DCAPS + d] = sS[tid * DCAPS + d] * inv;
      sP[tid] = scale;
    }
    __syncthreads();

    // ---- agreement update: b[n,c] += <u_hat[b,n,c,:], v[n,:]> ----
    if (iter < 2) {
      float vr[DCAPS];
#pragma unroll
      for (int d = 0; d < DCAPS; ++d) vr[d] = sV[n * DCAPS + d];

      for (int c = lane; c < ICAPS; c += 32) {
        const v4f* up = (const v4f*)(ub + (size_t)c * DCAPS);
        v4f q0 = up[0], q1 = up[1], q2 = up[2], q3 = up[3];
        float dot = 0.0f;
#pragma unroll
        for (int d = 0; d < 4; ++d) {
          dot += vr[d] * q0[d] + vr[d + 4] * q1[d]
               + vr[d + 8] * q2[d] + vr[d + 12] * q3[d];
        }
        Lg[n * ICAPS + c] += dot;
      }
      __syncthreads();
    }
  }

  // ---- outputs: v then p, concatenated flat ----
  if (tid < NCAPS * DCAPS) out_v[(size_t)b * NCAPS * DCAPS + tid] = sV[tid];
  if (tid < NCAPS)         out_p[(size_t)b * NCAPS + tid]         = sP[tid];
}

// ---------------------------------------------------------------------------
extern "C" void kernel_launch(void* const* d_in, const int* in_sizes, int n_in,
                              void* d_out, int out_size, void* d_ws, size_t ws_size,
                              hipStream_t stream) {
  (void)in_sizes; (void)n_in; (void)out_size; (void)ws_size;
  const float* x = (const float*)d_in[0];   // [32, 2048, 128]
  const float* W = (const float*)d_in[1];   // [32, 2048, 16, 128]
  float* uhat  = (float*)d_ws;              // [32, 32, 2048, 16] = 128 MB scratch
  float* out_v = (float*)d_out;             // [32, 32, 16]
  float* out_p = out_v + (size_t)B_SZ * NCAPS * DCAPS;  // [32, 32]

  dim3 g1(ICAPS / 64, NCAPS);
  uhat_wmma_kernel<<<g1, 256, 0, stream>>>(x, W, uhat);

  size_t smem = (size_t)(NCAPS * ICAPS + 2 * ICAPS + 2 * NCAPS * DCAPS + NCAPS) * sizeof(float);
  routing_kernel<<<dim3(B_SZ), 1024, smem, stream>>>(uhat, out_v, out_p);
}